// NodeDiscriminator_23003844838033
// MI455X (gfx1250) — compile-verified
//
#include <hip/hip_runtime.h>

// CDNA5 / gfx1250, wave32. Fused bilinear-score:
//   u = node @ W[0]    (fp32 WMMA 16x16x4)
//   out[i]     = dot(u[i], ctx[i])       + b
//   out[N + i] = dot(u[i], ctx_shift[i]) + b
// ctx_shift[0] = ctx[N-2], ctx_shift[i] = ctx[i-1]
//
// Fast path (needs ws >= 4MB): W transposed once into d_ws so B fragments are
// single b64 loads; A tile staged to LDS via async global->LDS (ASYNCcnt).

typedef __attribute__((ext_vector_type(2))) float v2f;
typedef __attribute__((ext_vector_type(8))) float v8f;
typedef __attribute__((ext_vector_type(4))) int   v4i;

#define NROWS 65536
#define HDIM  1024
#define TPW   8   // 16x16 N-tiles per wave; 8 waves * 8 * 16 = 1024 cols

#if __has_builtin(__builtin_amdgcn_global_load_async_to_lds_b128) && \
    __has_builtin(__builtin_amdgcn_s_wait_asynccnt)
#define USE_ASYNC_LDS 1
#endif

// ---------------- W transpose: WT[n*H + k] = W[k*H + n] ----------------
__global__ __launch_bounds__(256) void
transpose_w_kernel(const float* __restrict__ src, float* __restrict__ dst)
{
    __shared__ float tile[32][33];
    const int tx = threadIdx.x & 31;
    const int ty = threadIdx.x >> 5;            // 8 rows per pass
    const int bx = blockIdx.x * 32;             // k block
    const int by = blockIdx.y * 32;             // n block
#pragma unroll
    for (int r = ty; r < 32; r += 8)
        tile[r][tx] = src[(size_t)(bx + r) * HDIM + by + tx];
    __syncthreads();
#pragma unroll
    for (int r = ty; r < 32; r += 8)
        dst[(size_t)(by + r) * HDIM + bx + tx] = tile[tx][r];
}

// ---------------- Fast fused kernel (uses WT) ----------------
__global__ __launch_bounds__(256) void
node_disc_fast_kernel(const float* __restrict__ ctx,
                      const float* __restrict__ node,
                      const float* __restrict__ WT,   // [H][H], WT[n][k]
                      const float* __restrict__ bias,
                      float* __restrict__ out)
{
    extern __shared__ float lds_a[];   // 16 * 1024 floats = 64 KB
    __shared__ float s_pos[16];
    __shared__ float s_neg[16];

    const int tid  = threadIdx.x;
    const int wave = tid >> 5;
    const int lane = tid & 31;
    const int half = lane >> 4;
    const int l16  = lane & 15;
    const int m0   = blockIdx.x * 16;

    if (tid < 16) { s_pos[tid] = 0.0f; s_neg[tid] = 0.0f; }

    // ---- Stage A tile node[m0:m0+16, :] into LDS ----
    const float* asrc = node + (size_t)m0 * HDIM;
#ifdef USE_ASYNC_LDS
#pragma unroll
    for (int i = tid; i < (16 * HDIM) / 4; i += 256) {
        // global_load_async_to_lds_b128 (ASYNCcnt); generic ptrs implicitly
        // convert to the builtin's __device__/__shared__ AS params.
        __builtin_amdgcn_global_load_async_to_lds_b128(
            (v4i*)(asrc + i * 4), (v4i*)(lds_a + i * 4),
            /*offset=*/0, /*cpol=*/0);
    }
    __builtin_amdgcn_s_wait_asynccnt(0);
#else
    for (int i = tid; i < (16 * HDIM) / 4; i += 256) {
        float4 v = *(const float4*)(asrc + i * 4);
        *(float4*)(lds_a + i * 4) = v;
    }
#endif
    __syncthreads();

    // ---- GEMM tile: u[m0:m0+16, wave*128 .. +128] ----
    v8f acc[TPW] = {};

    // A frag: lanes 0-15 row M=l16 K={k,k+1}; lanes 16-31 K={k+2,k+3}
    const float* aof = lds_a + l16 * HDIM + 2 * half;
    // B frag: VGPR0 = W[k+2h][n] = WT[n][k+2h]; VGPR1 = WT[n][k+2h+1] (contig)
    const int n0 = wave * (TPW * 16) + l16;

    for (int k = 0; k < HDIM; k += 4) {
        v2f a = *(const v2f*)(aof + k);   // ds_load_b64
#pragma unroll
        for (int t = 0; t < TPW; ++t) {
            v2f b = *(const v2f*)(WT + (size_t)(n0 + t * 16) * HDIM
                                     + k + 2 * half);  // global_load_b64
            acc[t] = __builtin_amdgcn_wmma_f32_16x16x4_f32(
                false, a, false, b, (short)0, acc[t], false, false);
        }
    }

    // ---- Fused row-dot epilogue ----
    const float fb = bias[0];
#pragma unroll
    for (int vr = 0; vr < 8; ++vr) {
        const int r    = vr + 8 * half;
        const int g    = m0 + r;
        const int gneg = (g == 0) ? (NROWS - 2) : (g - 1);
        float p = 0.0f, q = 0.0f;
#pragma unroll
        for (int t = 0; t < TPW; ++t) {
            const float uval = acc[t][vr];
            const size_t n = (size_t)(n0 + t * 16);
            p += uval * ctx[(size_t)g    * HDIM + n];
            q += uval * ctx[(size_t)gneg * HDIM + n];
        }
        atomicAdd(&s_pos[r], p);
        atomicAdd(&s_neg[r], q);
    }
    __syncthreads();

    if (tid < 16) {
        out[m0 + tid] = s_pos[tid] + fb;
    } else if (tid < 32) {
        out[NROWS + m0 + (tid - 16)] = s_neg[tid - 16] + fb;
    }
}

// ---------------- Fallback kernel (no workspace): B from row-major W ----------------
__global__ __launch_bounds__(256) void
node_disc_fused_kernel(const float* __restrict__ ctx,
                       const float* __restrict__ node,
                       const float* __restrict__ W,
                       const float* __restrict__ bias,
                       float* __restrict__ out)
{
    __shared__ float s_pos[16];
    __shared__ float s_neg[16];

    const int tid  = threadIdx.x;
    const int wave = tid >> 5;
    const int lane = tid & 31;
    const int half = lane >> 4;
    const int l16  = lane & 15;
    const int m0   = blockIdx.x * 16;

    if (tid < 16) { s_pos[tid] = 0.0f; s_neg[tid] = 0.0f; }
    __syncthreads();

    v8f acc[TPW] = {};
    const float* arow  = node + (size_t)(m0 + l16) * HDIM + 2 * half;
    const int    n0    = wave * (TPW * 16) + l16;

    for (int k = 0; k < HDIM; k += 4) {
        v2f a = *(const v2f*)(arow + k);
        const float* wrow = W + (size_t)(k + 2 * half) * HDIM + n0;
#pragma unroll
        for (int t = 0; t < TPW; ++t) {
            v2f b;
            b.x = wrow[t * 16];
            b.y = wrow[t * 16 + HDIM];
            acc[t] = __builtin_amdgcn_wmma_f32_16x16x4_f32(
                false, a, false, b, (short)0, acc[t], false, false);
        }
    }

    const float fb = bias[0];
#pragma unroll
    for (int vr = 0; vr < 8; ++vr) {
        const int r    = vr + 8 * half;
        const int g    = m0 + r;
        const int gneg = (g == 0) ? (NROWS - 2) : (g - 1);
        float p = 0.0f, q = 0.0f;
#pragma unroll
        for (int t = 0; t < TPW; ++t) {
            const float uval = acc[t][vr];
            const size_t n = (size_t)(n0 + t * 16);
            p += uval * ctx[(size_t)g    * HDIM + n];
            q += uval * ctx[(size_t)gneg * HDIM + n];
        }
        atomicAdd(&s_pos[r], p);
        atomicAdd(&s_neg[r], q);
    }
    __syncthreads();

    if (tid < 16) {
        out[m0 + tid] = s_pos[tid] + fb;
    } else if (tid < 32) {
        out[NROWS + m0 + (tid - 16)] = s_neg[tid - 16] + fb;
    }
}

extern "C" void kernel_launch(void* const* d_in, const int* in_sizes, int n_in,
                              void* d_out, int out_size, void* d_ws, size_t ws_size,
                              hipStream_t stream) {
    const float* ctx  = (const float*)d_in[0];
    const float* node = (const float*)d_in[1];
    const float* W    = (const float*)d_in[2];
    const float* b    = (const float*)d_in[3];
    float* out = (float*)d_out;

    const size_t wt_bytes = (size_t)HDIM * HDIM * sizeof(float);  // 4 MB
    if (ws_size >= wt_bytes) {
        float* WT = (float*)d_ws;
        transpose_w_kernel<<<dim3(HDIM / 32, HDIM / 32), 256, 0, stream>>>(W, WT);
        node_disc_fast_kernel<<<dim3(NROWS / 16), 256,
                                16 * HDIM * sizeof(float), stream>>>(
            ctx, node, WT, b, out);
    } else {
        node_disc_fused_kernel<<<dim3(NROWS / 16), 256, 0, stream>>>(
            ctx, node, W, b, out);
    }
}